// PairwiseRankingLoss_3427383902894
// MI455X (gfx1250) — compile-verified
//
#include <hip/hip_runtime.h>
#include <limits.h>

typedef float v2f __attribute__((ext_vector_type(2)));
typedef float v8f __attribute__((ext_vector_type(8)));

#define H     64
#define WPB   8            // waves per block (wave32)
#define BLOCK (WPB * 32)

// One wave per race. diff[i][j] = s_i - s_j computed as rank-2 matmul via
// V_WMMA_F32_16X16X4_F32; hinge + pair-mask on VALU; pair COUNT via
// ballot + popcount (SALU, co-issues with VALU).
__global__ __launch_bounds__(BLOCK)
void pairwise_rank_loss_kernel(const float* __restrict__ scores,
                               const int* __restrict__ rankings,
                               const unsigned char* __restrict__ mask,
                               float* __restrict__ partials,
                               int B)
{
    __shared__ float s_sc[WPB][H];
    __shared__ int   s_kr[WPB][H];   // row key: valid ? rank : INT_MAX
    __shared__ int   s_kc[WPB][H];   // col key: valid ? rank : INT_MIN
    __shared__ float s_red[WPB][2];

    const int tid   = threadIdx.x;
    const int lane  = tid & 31;
    const int wave  = tid >> 5;
    const int batch = blockIdx.x * WPB + wave;
    const bool active = (batch < B);   // wave-uniform branch: EXEC all-1s inside

    if (active) {
        const long base = (long)batch * H;
#pragma unroll
        for (int t = 0; t < 2; ++t) {
            const int e = lane + t * 32;
            const float s = scores[base + e];
            const int   r = rankings[base + e];
            const bool valid = (mask[base + e] != 0) && (r > 0);
            s_sc[wave][e] = s;
            s_kr[wave][e] = valid ? r : INT_MAX;
            s_kc[wave][e] = valid ? r : INT_MIN;
        }
    }
    __syncthreads();

    float        loss = 0.0f;  // per-lane partial (reduced via shfl)
    unsigned int cntw = 0;     // wave-total pair count (uniform, SALU)

    if (active) {
        const float* sc  = s_sc[wave];
        const int*   krp = s_kr[wave];
        const int*   kcp = s_kc[wave];
        const bool lo  = (lane < 16);
        const int  l15 = lane & 15;

#pragma unroll
        for (int I = 0; I < 4; ++I) {
            // A fragment (16x4): lanes 0-15 hold row M=lane, K0=s_i, K1=1; K2,K3 = 0
            const float sA = sc[I * 16 + l15];
            v2f A;
            A.x = lo ? sA   : 0.0f;
            A.y = lo ? 1.0f : 0.0f;

            // 8 row keys for this lane's D rows (rows v / v+8)
            int kr[8];
            const int rbase = I * 16 + (lo ? 0 : 8);
#pragma unroll
            for (int v = 0; v < 8; ++v) kr[v] = krp[rbase + v];

#pragma unroll
            for (int J = 0; J < 4; ++J) {
                // B fragment (4x16): row K0 = ones, row K1 = -s_j; K2,K3 = 0
                const float sB = sc[J * 16 + l15];
                v2f Bf;
                Bf.x = lo ? 1.0f : 0.0f;
                Bf.y = lo ? -sB  : 0.0f;

                const int kc = kcp[J * 16 + l15];

                v8f c = {};
                // D[i][j] = s_i*1 + 1*(-s_j) = s_i - s_j
                v8f d = __builtin_amdgcn_wmma_f32_16x16x4_f32(
                    /*neg_a=*/false, A, /*neg_b=*/false, Bf,
                    /*c_mod=*/(short)0, c, /*reuse_a=*/false, /*reuse_b=*/false);

#pragma unroll
                for (int v = 0; v < 8; ++v) {
                    const bool  p = (kr[v] < kc);             // pair predicate
                    const float h = fmaxf(1.0f - d[v], 0.0f); // hinge
                    loss += p ? h : 0.0f;                     // VALU
                    // wave-total count: v_cmp mask -> s_bcnt (SALU, co-issued)
                    cntw += __builtin_popcount(__builtin_amdgcn_ballot_w32(p));
                }
            }
        }
    }

    // wave32 reduction of loss only (count is already wave-uniform & exact)
#pragma unroll
    for (int off = 16; off > 0; off >>= 1) {
        loss += __shfl_xor(loss, off, 32);
    }
    if (lane == 0) {
        const float cnt = (float)cntw;
        const float bl  = (cntw > 0u) ? (loss / cnt) : 0.0f;
        const float has = (cntw > 0u) ? 1.0f : 0.0f;
        s_red[wave][0] = bl;
        s_red[wave][1] = has;
    }
    __syncthreads();

    if (tid == 0) {
        float t0 = 0.0f, t1 = 0.0f;
#pragma unroll
        for (int w = 0; w < WPB; ++w) { t0 += s_red[w][0]; t1 += s_red[w][1]; }
        partials[2 * blockIdx.x + 0] = t0;   // sum of per-race losses
        partials[2 * blockIdx.x + 1] = t1;   // number of contributing races
    }
}

// Deterministic fixed-order final reduction (no float atomics).
__global__ __launch_bounds__(256)
void pairwise_rank_loss_finalize(const float* __restrict__ partials,
                                 float* __restrict__ out,
                                 int nblocks)
{
    float t = 0.0f, n = 0.0f;
    for (int i = threadIdx.x; i < nblocks; i += 256) {
        t += partials[2 * i + 0];
        n += partials[2 * i + 1];
    }
#pragma unroll
    for (int off = 16; off > 0; off >>= 1) {
        t += __shfl_xor(t, off, 32);
        n += __shfl_xor(n, off, 32);
    }
    __shared__ float st[8], sn[8];
    const int wave = threadIdx.x >> 5;
    const int lane = threadIdx.x & 31;
    if (lane == 0) { st[wave] = t; sn[wave] = n; }
    __syncthreads();
    if (threadIdx.x == 0) {
        float T = 0.0f, N = 0.0f;
#pragma unroll
        for (int w = 0; w < 8; ++w) { T += st[w]; N += sn[w]; }
        out[0] = (N > 0.0f) ? (T / N) : 0.0f;
    }
}

extern "C" void kernel_launch(void* const* d_in, const int* in_sizes, int n_in,
                              void* d_out, int out_size, void* d_ws, size_t ws_size,
                              hipStream_t stream)
{
    const float*         scores   = (const float*)d_in[0];
    const int*           rankings = (const int*)d_in[1];
    const unsigned char* maskp    = (const unsigned char*)d_in[2];

    const int B       = in_sizes[0] / H;                  // 16384 races
    const int nblocks = (B + WPB - 1) / WPB;              // 2048 blocks
    float* partials   = (float*)d_ws;                     // 2*nblocks floats (16 KB)

    pairwise_rank_loss_kernel<<<nblocks, BLOCK, 0, stream>>>(
        scores, rankings, maskp, partials, B);
    pairwise_rank_loss_finalize<<<1, 256, 0, stream>>>(
        partials, (float*)d_out, nblocks);
}